// InvariantPointAttention_58213986730607
// MI455X (gfx1250) — compile-verified
//
#include <hip/hip_runtime.h>
#include <hip/hip_bf16.h>
#include <math.h>

typedef float v2f __attribute__((ext_vector_type(2)));
typedef float v8f __attribute__((ext_vector_type(8)));

#define N_RES   512
#define CS      384
#define CZ      128
#define CH      16
#define NH      12
#define PQv     4
#define PVv     8
#define QKV_W   576      // 3*NH*CH
#define PTS_W   288      // 2*NH*PQ*3
#define VPTS_W  288      // NH*PV*3
#define CONCATW 2112
#define W_C_HALF 0.11785113019775792f  // sqrt(2/(9*4))/2
#define W_L      0.5773502691896258f   // sqrt(1/3)

__device__ __forceinline__ v8f wmma4(v2f a, v2f b, v8f c) {
  // V_WMMA_F32_16X16X4_F32 : D(16x16 f32) = A(16x4) * B(4x16) + C
  return __builtin_amdgcn_wmma_f32_16x16x4_f32(false, a, false, b, (short)0, c,
                                               false, false);
}

// ---------------------------------------------------------------------------
// Generic fp32 WMMA GEMM: C[M,N] = A[M,K] @ B[K,N] (+ bias[N]); M%16==N%16==0,
// K%4==0. One wave per 16x16 output tile.
// ---------------------------------------------------------------------------
__global__ void gemm_f32_wmma(const float* __restrict__ A,
                              const float* __restrict__ B,
                              const float* __restrict__ bias,
                              float* __restrict__ C, int M, int N, int K) {
  int wave = (int)((blockIdx.x * blockDim.x + threadIdx.x) >> 5);
  int lane = threadIdx.x & 31;
  int ntiles = N >> 4;
  int mt = wave / ntiles, nt = wave - mt * ntiles;
  if (mt >= (M >> 4)) return;
  int lo = lane & 15, hi = lane >> 4;
  int arow = mt * 16 + lo;
  int bcol = nt * 16 + lo;
  v8f acc = {};
  for (int k = 0; k < K; k += 4) {
    int kk = k + 2 * hi;
    v2f a, b;
    a.x = A[arow * K + kk];
    a.y = A[arow * K + kk + 1];
    b.x = B[kk * N + bcol];
    b.y = B[(kk + 1) * N + bcol];
    acc = wmma4(a, b, acc);
  }
  float bv = bias ? bias[bcol] : 0.0f;
  for (int r = 0; r < 8; ++r) {
    int row = mt * 16 + r + 8 * hi;
    C[row * N + bcol] = acc[r] + bv;
  }
}

// ---------------------------------------------------------------------------
// Rigid transforms: qg/kg from qk_pts proj, vg from v_pts proj.
// qg,kg: [n][NH][PQ][3] = [n][144]; vg: [n][NH][PV][3] = [n][288]
// ---------------------------------------------------------------------------
__global__ void rigid_fwd_kernel(const float* __restrict__ qkp,
                                 const float* __restrict__ vpp,
                                 const float* __restrict__ R,
                                 const float* __restrict__ t,
                                 float* __restrict__ qg, float* __restrict__ kg,
                                 float* __restrict__ vg) {
  int idx = blockIdx.x * blockDim.x + threadIdx.x;
  int total_qk = N_RES * 96;  // 2*NH*PQ points per residue
  float px, py, pz;
  const float* Rn;
  const float* tn;
  float* dst;
  if (idx < total_qk) {
    int n = idx / 96, rem = idx - n * 96;  // rem = sel*48 + h*4 + p
    const float* src = qkp + n * PTS_W + rem * 3;
    px = src[0]; py = src[1]; pz = src[2];
    Rn = R + n * 9; tn = t + n * 3;
    dst = (rem < 48) ? (qg + n * 144 + rem * 3) : (kg + n * 144 + (rem - 48) * 3);
  } else {
    int idx2 = idx - total_qk;
    if (idx2 >= N_RES * 96) return;     // NH*PV points per residue
    int n = idx2 / 96, rem = idx2 - n * 96;
    const float* src = vpp + n * VPTS_W + rem * 3;
    px = src[0]; py = src[1]; pz = src[2];
    Rn = R + n * 9; tn = t + n * 3;
    dst = vg + n * VPTS_W + rem * 3;
  }
  for (int x = 0; x < 3; ++x)
    dst[x] = Rn[x * 3 + 0] * px + Rn[x * 3 + 1] * py + Rn[x * 3 + 2] * pz + tn[x];
}

// qn/kn[n][NH]: squared norms over PQ*3
__global__ void pt_norm_kernel(const float* __restrict__ qg,
                               const float* __restrict__ kg,
                               float* __restrict__ qn, float* __restrict__ kn) {
  int idx = blockIdx.x * blockDim.x + threadIdx.x;
  if (idx >= 2 * N_RES * NH) return;
  int sel = idx / (N_RES * NH);
  int r = idx - sel * (N_RES * NH);
  int n = r / NH, h = r - n * NH;
  const float* src = (sel == 0 ? qg : kg) + n * 144 + h * 12;
  float s = 0.f;
  for (int k = 0; k < 12; ++k) s += src[k] * src[k];
  (sel == 0 ? qn : kn)[n * NH + h] = s;
}

// ---------------------------------------------------------------------------
// term2: logits[h][i][j] = (z[i,j,:] @ Wb)[h] + bb[h].  WMMA over flattened
// (i,j) rows (M = N*N), K=128, 16 cols (12 heads + pad). LDS transpose so the
// store into [h][i][j] planes is contiguous in j.
// ---------------------------------------------------------------------------
__global__ void term2_kernel(const float* __restrict__ z,
                             const float* __restrict__ Wb,
                             const float* __restrict__ bb,
                             float* __restrict__ logits) {
  __shared__ float lds[8][16][17];
  int w = threadIdx.x >> 5;
  int wave = (int)((blockIdx.x * blockDim.x + threadIdx.x) >> 5);
  int lane = threadIdx.x & 31;
  int lo = lane & 15, hi = lane >> 4;
  int m0 = wave * 16;                 // 16 consecutive (i,j) flat rows
  int arow = m0 + lo;
  v8f acc = {};
  for (int k = 0; k < CZ; k += 4) {
    int kk = k + 2 * hi;
    v2f a, b;
    a.x = z[arow * CZ + kk];
    a.y = z[arow * CZ + kk + 1];
    b.x = (lo < NH) ? Wb[kk * NH + lo] : 0.0f;
    b.y = (lo < NH) ? Wb[(kk + 1) * NH + lo] : 0.0f;
    acc = wmma4(a, b, acc);
  }
  for (int r = 0; r < 8; ++r) lds[w][r + 8 * hi][lo] = acc[r];
  __syncthreads();
  int i = m0 >> 9, j0 = m0 & 511;
  for (int hh = 0; hh < 6; ++hh) {
    int h = hh * 2 + hi;                         // 0..11
    float v = lds[w][lo][h] + bb[h];             // row=j_local, col=h
    logits[(h * N_RES + i) * N_RES + j0 + lo] = v;
  }
}

// ---------------------------------------------------------------------------
// logits[h][i][j] = W_L*(qk/4 + term2 - softplus(g)*W_C/2*(qn+kn-2*ptdot)),
// masked.  One wave per (h, 16x16 tile).
// ---------------------------------------------------------------------------
__global__ void logits_kernel(const float* __restrict__ qkv,
                              const float* __restrict__ qg,
                              const float* __restrict__ kg,
                              const float* __restrict__ qn,
                              const float* __restrict__ kn,
                              const float* __restrict__ gamma,
                              const unsigned char* __restrict__ mask,
                              float* __restrict__ logits) {
  int wave = (int)((blockIdx.x * blockDim.x + threadIdx.x) >> 5);
  int lane = threadIdx.x & 31;
  int lo = lane & 15, hi = lane >> 4;
  int jt = wave & 31, it = (wave >> 5) & 31, h = wave >> 10;  // 12*32*32 waves
  int i0 = it * 16, j0 = jt * 16;

  v8f acc1 = {};  // q . k  (K = 16)
  const float* qb = qkv + (i0 + lo) * QKV_W + h * CH;            // q row
  const float* kb = qkv + (j0 + lo) * QKV_W + NH * CH + h * CH;  // k row (B^T col)
  for (int k = 0; k < CH; k += 4) {
    int kk = k + 2 * hi;
    v2f a, b;
    a.x = qb[kk]; a.y = qb[kk + 1];
    b.x = kb[kk]; b.y = kb[kk + 1];
    acc1 = wmma4(a, b, acc1);
  }
  v8f acc2 = {};  // qg . kg (K = 12)
  const float* qgb = qg + (i0 + lo) * 144 + h * 12;
  const float* kgb = kg + (j0 + lo) * 144 + h * 12;
  for (int k = 0; k < 12; k += 4) {
    int kk = k + 2 * hi;
    v2f a, b;
    a.x = qgb[kk]; a.y = qgb[kk + 1];
    b.x = kgb[kk]; b.y = kgb[kk + 1];
    acc2 = wmma4(a, b, acc2);
  }
  float c3 = log1pf(expf(gamma[h])) * W_C_HALF;
  float knj = kn[(j0 + lo) * NH + h];
  bool mj = mask[j0 + lo] != 0;
  for (int r = 0; r < 8; ++r) {
    int i = i0 + r + 8 * hi;
    int j = j0 + lo;
    float t1 = acc1[r] * 0.25f;  // 1/sqrt(CH)
    float d2 = qn[i * NH + h] + knj - 2.0f * acc2[r];
    int idx = (h * N_RES + i) * N_RES + j;
    float lg = W_L * (t1 + logits[idx] - c3 * d2);
    logits[idx] = mj ? lg : -1.0e9f;
  }
}

// ---------------------------------------------------------------------------
// Row softmax over j: one 256-thread block per (h,i) row of 512.
// ---------------------------------------------------------------------------
__global__ void softmax_kernel(float* __restrict__ logits) {
  float* p = logits + (long)blockIdx.x * N_RES;
  int tid = threadIdx.x;
  __shared__ float red[8];
  float v0 = p[tid], v1 = p[tid + 256];
  float m = fmaxf(v0, v1);
  for (int off = 16; off > 0; off >>= 1) m = fmaxf(m, __shfl_xor(m, off, 32));
  if ((tid & 31) == 0) red[tid >> 5] = m;
  __syncthreads();
  float bm = red[0];
  for (int i = 1; i < 8; ++i) bm = fmaxf(bm, red[i]);
  __syncthreads();
  float e0 = expf(v0 - bm), e1 = expf(v1 - bm);
  float s = e0 + e1;
  for (int off = 16; off > 0; off >>= 1) s += __shfl_xor(s, off, 32);
  if ((tid & 31) == 0) red[tid >> 5] = s;
  __syncthreads();
  float bs = red[0];
  for (int i = 1; i < 8; ++i) bs += red[i];
  float inv = 1.0f / bs;
  p[tid] = e0 * inv;
  p[tid + 256] = e1 * inv;
}

// ---------------------------------------------------------------------------
// out_tilde[i,h,c] = sum_j aw[h,i,j] z[i,j,c] -> feats[i][h*128+c]
// One wave per (i, 16-col tile of CZ). A rows = heads (pad to 16), K = 512.
// ---------------------------------------------------------------------------
__global__ void out_tilde_kernel(const float* __restrict__ aw,
                                 const float* __restrict__ z,
                                 float* __restrict__ feats) {
  int wave = (int)((blockIdx.x * blockDim.x + threadIdx.x) >> 5);
  int lane = threadIdx.x & 31;
  int lo = lane & 15, hi = lane >> 4;
  int i = wave >> 3, nt = wave & 7;
  int n0 = nt * 16;
  v8f acc = {};
  for (int k = 0; k < N_RES; k += 4) {
    int kk = k + 2 * hi;
    v2f a, b;
    a.x = (lo < NH) ? aw[(lo * N_RES + i) * N_RES + kk] : 0.0f;
    a.y = (lo < NH) ? aw[(lo * N_RES + i) * N_RES + kk + 1] : 0.0f;
    b.x = z[(i * N_RES + kk) * CZ + n0 + lo];
    b.y = z[(i * N_RES + kk + 1) * CZ + n0 + lo];
    acc = wmma4(a, b, acc);
  }
  for (int r = 0; r < 8; ++r) {
    int h = r + 8 * hi;
    if (h < NH) feats[i * CONCATW + h * CZ + n0 + lo] = acc[r];
  }
}

// out[i,h,c] = sum_j aw[h,i,j] v[j,h,c] -> feats[i][1536 + h*16 + c]
__global__ void out_v_kernel(const float* __restrict__ aw,
                             const float* __restrict__ qkv,
                             float* __restrict__ feats) {
  int wave = (int)((blockIdx.x * blockDim.x + threadIdx.x) >> 5);
  int lane = threadIdx.x & 31;
  int lo = lane & 15, hi = lane >> 4;
  int h = wave >> 5, it = wave & 31;
  int i0 = it * 16;
  v8f acc = {};
  const float* awh = aw + (long)h * N_RES * N_RES;
  for (int k = 0; k < N_RES; k += 4) {
    int kk = k + 2 * hi;
    v2f a, b;
    a.x = awh[(i0 + lo) * N_RES + kk];
    a.y = awh[(i0 + lo) * N_RES + kk + 1];
    b.x = qkv[kk * QKV_W + 2 * NH * CH + h * CH + lo];
    b.y = qkv[(kk + 1) * QKV_W + 2 * NH * CH + h * CH + lo];
    acc = wmma4(a, b, acc);
  }
  for (int r = 0; r < 8; ++r) {
    int i = i0 + r + 8 * hi;
    feats[i * CONCATW + NH * CZ + h * CH + lo] = acc[r];
  }
}

// o_global[i,h,n] = sum_j aw[h,i,j] vg[j,h,n], n in [0,24)
__global__ void out_pts_kernel(const float* __restrict__ aw,
                               const float* __restrict__ vg,
                               float* __restrict__ oglob) {
  int wave = (int)((blockIdx.x * blockDim.x + threadIdx.x) >> 5);
  int lane = threadIdx.x & 31;
  int lo = lane & 15, hi = lane >> 4;
  int h = wave >> 6, rem = wave & 63;
  int it = rem >> 1, nt = rem & 1;
  int i0 = it * 16, n0 = nt * 16;
  bool inb = (n0 + lo) < 24;
  v8f acc = {};
  const float* awh = aw + (long)h * N_RES * N_RES;
  for (int k = 0; k < N_RES; k += 4) {
    int kk = k + 2 * hi;
    v2f a, b;
    a.x = awh[(i0 + lo) * N_RES + kk];
    a.y = awh[(i0 + lo) * N_RES + kk + 1];
    b.x = inb ? vg[kk * VPTS_W + h * 24 + n0 + lo] : 0.0f;
    b.y = inb ? vg[(kk + 1) * VPTS_W + h * 24 + n0 + lo] : 0.0f;
    acc = wmma4(a, b, acc);
  }
  for (int r = 0; r < 8; ++r) {
    int i = i0 + r + 8 * hi;
    if (inb) oglob[i * VPTS_W + h * 24 + n0 + lo] = acc[r];
  }
}

// inverse rigid + norm -> feats[i][1728 + (h*8+v)*3 + x], feats[i][2016 + h*8+v]
__global__ void rigid_inv_kernel(const float* __restrict__ oglob,
                                 const float* __restrict__ R,
                                 const float* __restrict__ t,
                                 float* __restrict__ feats) {
  int idx = blockIdx.x * blockDim.x + threadIdx.x;
  if (idx >= N_RES * NH * PVv) return;
  int i = idx / 96, rem = idx - i * 96;  // rem = h*8 + v
  const float* g = oglob + i * VPTS_W + rem * 3;
  const float* Rn = R + i * 9;
  const float* tn = t + i * 3;
  float d0 = g[0] - tn[0], d1 = g[1] - tn[1], d2 = g[2] - tn[2];
  float o[3];
  for (int x = 0; x < 3; ++x)
    o[x] = Rn[0 * 3 + x] * d0 + Rn[1 * 3 + x] * d1 + Rn[2 * 3 + x] * d2;
  float* f = feats + i * CONCATW;
  f[1728 + rem * 3 + 0] = o[0];
  f[1728 + rem * 3 + 1] = o[1];
  f[1728 + rem * 3 + 2] = o[2];
  f[2016 + rem] = sqrtf(o[0] * o[0] + o[1] * o[1] + o[2] * o[2] + 1e-8f);
}

// ---------------------------------------------------------------------------
extern "C" void kernel_launch(void* const* d_in, const int* in_sizes, int n_in,
                              void* d_out, int out_size, void* d_ws,
                              size_t ws_size, hipStream_t stream) {
  const float* s        = (const float*)d_in[0];
  const float* z        = (const float*)d_in[1];
  const float* R        = (const float*)d_in[2];
  const float* t        = (const float*)d_in[3];
  const unsigned char* mask = (const unsigned char*)d_in[4];
  const float* Wqkv     = (const float*)d_in[5];
  const float* Wqk_pts  = (const float*)d_in[6];
  const float* bqk_pts  = (const float*)d_in[7];
  const float* Wv_pts   = (const float*)d_in[8];
  const float* bv_pts   = (const float*)d_in[9];
  const float* Wb       = (const float*)d_in[10];
  const float* bb       = (const float*)d_in[11];
  const float* gamma    = (const float*)d_in[12];
  const float* Wfin     = (const float*)d_in[13];
  const float* bfin     = (const float*)d_in[14];
  float* out = (float*)d_out;

  float* ws = (float*)d_ws;
  float* w_qkv   = ws;                 // 512*576   = 294912
  float* w_qkp   = w_qkv   + 294912;   // 512*288   = 147456
  float* w_vpp   = w_qkp   + 147456;   // 512*288   = 147456
  float* w_qg    = w_vpp   + 147456;   // 512*144   = 73728
  float* w_kg    = w_qg    + 73728;    // 512*144   = 73728
  float* w_vg    = w_kg    + 73728;    // 512*288   = 147456
  float* w_qn    = w_vg    + 147456;   // 512*12    = 6144
  float* w_kn    = w_qn    + 6144;     // 512*12    = 6144
  float* w_oglob = w_kn    + 6144;     // 512*288   = 147456
  float* w_feats = w_oglob + 147456;   // 512*2112  = 1081344
  float* w_logit = w_feats + 1081344;  // 12*512*512 = 3145728

  const int BT = 256;  // 8 waves per block

  // 1-3: projections from s
  gemm_f32_wmma<<<(32 * 36 + 7) / 8, BT, 0, stream>>>(s, Wqkv, nullptr, w_qkv,
                                                      N_RES, QKV_W, CS);
  gemm_f32_wmma<<<(32 * 18 + 7) / 8, BT, 0, stream>>>(s, Wqk_pts, bqk_pts,
                                                      w_qkp, N_RES, PTS_W, CS);
  gemm_f32_wmma<<<(32 * 18 + 7) / 8, BT, 0, stream>>>(s, Wv_pts, bv_pts, w_vpp,
                                                      N_RES, VPTS_W, CS);
  // 4: rigid transforms, 5: point norms
  rigid_fwd_kernel<<<(2 * N_RES * 96 + BT - 1) / BT, BT, 0, stream>>>(
      w_qkp, w_vpp, R, t, w_qg, w_kg, w_vg);
  pt_norm_kernel<<<(2 * N_RES * NH + BT - 1) / BT, BT, 0, stream>>>(w_qg, w_kg,
                                                                    w_qn, w_kn);
  // 6: term2 (first z pass) writes into logits planes
  term2_kernel<<<(N_RES * N_RES / 16) / 8, BT, 0, stream>>>(z, Wb, bb, w_logit);
  // 7: full logits
  logits_kernel<<<(NH * 32 * 32) / 8, BT, 0, stream>>>(w_qkv, w_qg, w_kg, w_qn,
                                                       w_kn, gamma, mask,
                                                       w_logit);
  // 8: softmax rows -> aw (in place)
  softmax_kernel<<<NH * N_RES, BT, 0, stream>>>(w_logit);
  // 9: out_tilde (second z pass)
  out_tilde_kernel<<<(N_RES * 8) / 8, BT, 0, stream>>>(w_logit, z, w_feats);
  // 10: out = aw @ v
  out_v_kernel<<<(NH * 32) / 8, BT, 0, stream>>>(w_logit, w_qkv, w_feats);
  // 11: o_global = aw @ vg
  out_pts_kernel<<<(NH * 32 * 2) / 8, BT, 0, stream>>>(w_logit, w_vg, w_oglob);
  // 12: inverse rigid + norms
  rigid_inv_kernel<<<(N_RES * 96 + BT - 1) / BT, BT, 0, stream>>>(w_oglob, R, t,
                                                                  w_feats);
  // 13: final projection
  gemm_f32_wmma<<<(32 * 24 + 7) / 8, BT, 0, stream>>>(w_feats, Wfin, bfin, out,
                                                      N_RES, CS, CONCATW);
}